// IntraAggr_31344671326263
// MI455X (gfx1250) — compile-verified
//
#include <hip/hip_runtime.h>

// MI455X / gfx1250 implementation.
// Design: edge-parallel gather + L2-resident f32 atomic scatter (166MB state < 192MB L2),
// wave32 16-row-stripe finalize using exact-f32 V_WMMA_F32_16X16X4_F32 for out += w*norm.
// All finalize accesses go through per-array base pointers (row0/lane/lane-half folded in)
// so every load/store is an immediate-offset clause member; B loads batch like C loads.

typedef __attribute__((ext_vector_type(2))) float v2f;
typedef __attribute__((ext_vector_type(8))) float v8f;

#define EMB  128
#define HALF 64

// ---------------- init: cur = out = emb ----------------
__global__ void init_copy_kernel(const float* __restrict__ emb,
                                 float* __restrict__ cur,
                                 float* __restrict__ out, int n4) {
  int i = blockIdx.x * blockDim.x + threadIdx.x;
  if (i >= n4) return;
  float4 v = ((const float4*)emb)[i];
  ((float4*)cur)[i] = v;
  ((float4*)out)[i] = v;
}

// ---------------- degree counting (once; degrees are layer-invariant) ----------------
__global__ void count_kernel(const int* __restrict__ dIdx, int E,
                             unsigned* __restrict__ cnt) {
  int e = blockIdx.x * blockDim.x + threadIdx.x;
  if (e >= E) return;
  atomicAdd(&cnt[dIdx[e]], 1u);
}

__global__ void inv_kernel(const unsigned* __restrict__ cnt,
                           float* __restrict__ inv, int n) {
  int i = blockIdx.x * blockDim.x + threadIdx.x;
  if (i >= n) return;
  inv[i] = 1.0f / fmaxf((float)cnt[i], 1.0f);
}

// ---------------- edge scatter: 16 lanes x float4 = one 256B row per edge ----------------
__global__ void scatter_kernel(const float* __restrict__ srcF, int srcCol,
                               float* __restrict__ dstF, int dstCol,
                               const int* __restrict__ sIdx,
                               const int* __restrict__ dIdx, int E) {
  int gid = blockIdx.x * blockDim.x + threadIdx.x;
  int e = gid >> 4;
  if (e >= E) return;
  int c = (gid & 15) << 2;            // 4 floats per lane, 16 lanes cover 64 cols
  int s = sIdx[e];
  int d = dIdx[e];
  float4 v = *(const float4*)(srcF + (long long)s * EMB + srcCol + c);
  float* dp = dstF + (long long)d * EMB + dstCol + c;
  atomicAdd(dp + 0, v.x);
  atomicAdd(dp + 1, v.y);
  atomicAdd(dp + 2, v.z);
  atomicAdd(dp + 3, v.w);
}

// ---------------- finalize (full 16-row stripes, no guards) ----------------
// Each wave32 owns 16 full rows. Per 16x16 column tile, 4 chained
// V_WMMA_F32_16X16X4_F32 with A = w * 4-row identity-selection blocks compute
// D = C + w * Norm exactly in f32 (one nonzero product per output element).
__global__ void finalize_full_kernel(const float* __restrict__ acc,
                                     float* __restrict__ cur,
                                     float* __restrict__ out,
                                     const float* __restrict__ invL,
                                     const float* __restrict__ invR,
                                     int nStripes, float w) {
  int wave = (int)((blockIdx.x * blockDim.x + threadIdx.x) >> 5);
  int lane = threadIdx.x & 31;
  if (wave >= nStripes) return;
  long long row0 = (long long)wave * 16;
  int h = lane >> 4;    // lane half (K/M split per ISA layout tables)
  int l = lane & 15;

  // Hoist per-row reciprocals and A selection constants out of the tile loop.
  // Lane touches rows r(k,v) = 4k + 2h + v of the stripe for its B chunks.
  float ivL[8], ivR[8];
  v2f a[4];
#pragma unroll
  for (int k = 0; k < 4; ++k) {
#pragma unroll
    for (int v = 0; v < 2; ++v) {
      long long grow = row0 + 4 * k + 2 * h + v;
      ivL[k * 2 + v] = invL[grow];
      ivR[k * 2 + v] = invR[grow];
      a[k][v] = (l == 4 * k + 2 * h + v) ? w : 0.0f;   // A[M=l][K=2h+v] of chunk k
    }
  }

  // Base pointers with all runtime terms (row0, lane, lane-half) folded in:
  // every subsequent access offset is a compile-time constant -> clause + imm offset.
  const float* __restrict__ bAcc = acc + row0 * EMB + 2 * h * EMB + l;  // B rows: +(4k+v)*EMB + t*16
  float*       __restrict__ bCur = cur + row0 * EMB + 2 * h * EMB + l;
  const float* __restrict__ cOutL = out + row0 * EMB + 8 * h * EMB + l; // C/D rows: +r*EMB + t*16
  float*       __restrict__ cOutS = out + row0 * EMB + 8 * h * EMB + l;

#pragma unroll
  for (int t = 0; t < 8; ++t) {       // 8 column tiles of 16 cover EMB=128
    // ---- C tile: one clause of 8 imm-offset loads ----
    v8f c;
#pragma unroll
    for (int r = 0; r < 8; ++r)
      c[r] = cOutL[r * EMB + t * 16];

    // ---- B tile: one clause of 8 imm-offset loads, then muls + cur stores ----
    float raw[8];
#pragma unroll
    for (int k = 0; k < 4; ++k)
#pragma unroll
      for (int v = 0; v < 2; ++v)
        raw[k * 2 + v] = bAcc[(4 * k + v) * EMB + t * 16];

    v2f b[4];
#pragma unroll
    for (int k = 0; k < 4; ++k) {
#pragma unroll
      for (int v = 0; v < 2; ++v) {
        float av = raw[k * 2 + v] * (t < 4 ? ivL[k * 2 + v] : ivR[k * 2 + v]);
        bCur[(4 * k + v) * EMB + t * 16] = av;   // next-layer features (each element once)
        b[k][v] = av;
      }
    }

    // ---- exact f32 axpy on the matrix pipe ----
#pragma unroll
    for (int k = 0; k < 4; ++k)
      c = __builtin_amdgcn_wmma_f32_16x16x4_f32(false, a[k], false, b[k],
                                                (short)0, c, false, false);

    // ---- D tile: one clause of 8 imm-offset stores ----
#pragma unroll
    for (int r = 0; r < 8; ++r)
      cOutS[r * EMB + t * 16] = c[r];
  }
}

// ---------------- finalize tail (<16 rows, elementwise) ----------------
__global__ void finalize_tail_kernel(const float* __restrict__ acc,
                                     float* __restrict__ cur,
                                     float* __restrict__ out,
                                     const float* __restrict__ invL,
                                     const float* __restrict__ invR,
                                     int rowStart, int nRows, float w) {
  int idx = blockIdx.x * blockDim.x + threadIdx.x;
  int n = (nRows - rowStart) * EMB;
  if (idx >= n) return;
  int row = rowStart + idx / EMB;
  int col = idx % EMB;
  float iv = (col < HALF) ? invL[row] : invR[row];
  float av = acc[(long long)row * EMB + col] * iv;
  cur[(long long)row * EMB + col] = av;
  out[(long long)row * EMB + col] += w * av;
}

extern "C" void kernel_launch(void* const* d_in, const int* in_sizes, int n_in,
                              void* d_out, int out_size, void* d_ws, size_t ws_size,
                              hipStream_t stream) {
  const float* uE = (const float*)d_in[0];
  const float* vE = (const float*)d_in[1];
  const float* pE = (const float*)d_in[2];
  const float* tE = (const float*)d_in[3];
  const int* uv_s = (const int*)d_in[4];  const int* uv_d = (const int*)d_in[5];
  const int* vu_s = (const int*)d_in[6];  const int* vu_d = (const int*)d_in[7];
  const int* up_s = (const int*)d_in[8];  const int* up_d = (const int*)d_in[9];
  const int* pu_s = (const int*)d_in[10]; const int* pu_d = (const int*)d_in[11];
  const int* vt_s = (const int*)d_in[12]; const int* vt_d = (const int*)d_in[13];
  const int* tv_s = (const int*)d_in[14]; const int* tv_d = (const int*)d_in[15];
  const int* pt_s = (const int*)d_in[16]; const int* pt_d = (const int*)d_in[17];
  const int* tp_s = (const int*)d_in[18]; const int* tp_d = (const int*)d_in[19];

  const int NU = in_sizes[0] / EMB, NV = in_sizes[1] / EMB;
  const int NP = in_sizes[2] / EMB, NT = in_sizes[3] / EMB;
  const int E_uv = in_sizes[4], E_vu = in_sizes[6], E_up = in_sizes[8],  E_pu = in_sizes[10];
  const int E_vt = in_sizes[12], E_tv = in_sizes[14], E_pt = in_sizes[16], E_tp = in_sizes[18];

  const long long ROWS = (long long)NU + NV + NP + NT;
  const long long FEAT = ROWS * EMB;

  // workspace layout: cur | acc | inv-degree | counts  (~338 MB)
  float*    cur = (float*)d_ws;
  float*    acc = cur + FEAT;
  float*    inv = acc + FEAT;
  unsigned* cnt = (unsigned*)(inv + 2 * ROWS);

  float* out = (float*)d_out;

  // row offsets per node type (outputs concatenated in return order)
  const long long uOff = 0, vOff = NU, pOff = (long long)NU + NV, tOff = (long long)NU + NV + NP;
  // inverse-degree table offsets per (relation -> dst) pair
  const long long o_vu = 0,            o_pu = NU;
  const long long o_uv = 2LL * NU,     o_tv = 2LL * NU + NV;
  const long long o_up = 2LL * NU + 2LL * NV, o_tp = o_up + NP;
  const long long o_vt = o_tp + NP,    o_pt = o_vt + NT;

  // ---- init: out = cur = embeddings ----
  init_copy_kernel<<<(NU * 32 + 255) / 256, 256, 0, stream>>>(uE, cur + uOff * EMB, out + uOff * EMB, NU * 32);
  init_copy_kernel<<<(NV * 32 + 255) / 256, 256, 0, stream>>>(vE, cur + vOff * EMB, out + vOff * EMB, NV * 32);
  init_copy_kernel<<<(NP * 32 + 255) / 256, 256, 0, stream>>>(pE, cur + pOff * EMB, out + pOff * EMB, NP * 32);
  init_copy_kernel<<<(NT * 32 + 255) / 256, 256, 0, stream>>>(tE, cur + tOff * EMB, out + tOff * EMB, NT * 32);

  // ---- degrees (layer-invariant) ----
  hipMemsetAsync(cnt, 0, (size_t)(2 * ROWS) * sizeof(unsigned), stream);
  count_kernel<<<(E_vu + 255) / 256, 256, 0, stream>>>(vu_d, E_vu, cnt + o_vu);
  count_kernel<<<(E_pu + 255) / 256, 256, 0, stream>>>(pu_d, E_pu, cnt + o_pu);
  count_kernel<<<(E_uv + 255) / 256, 256, 0, stream>>>(uv_d, E_uv, cnt + o_uv);
  count_kernel<<<(E_tv + 255) / 256, 256, 0, stream>>>(tv_d, E_tv, cnt + o_tv);
  count_kernel<<<(E_up + 255) / 256, 256, 0, stream>>>(up_d, E_up, cnt + o_up);
  count_kernel<<<(E_tp + 255) / 256, 256, 0, stream>>>(tp_d, E_tp, cnt + o_tp);
  count_kernel<<<(E_vt + 255) / 256, 256, 0, stream>>>(vt_d, E_vt, cnt + o_vt);
  count_kernel<<<(E_pt + 255) / 256, 256, 0, stream>>>(pt_d, E_pt, cnt + o_pt);
  inv_kernel<<<(int)((2 * ROWS + 255) / 256), 256, 0, stream>>>(cnt, inv, (int)(2 * ROWS));

  // ---- 3 layers ----
  for (int i = 0; i < 3; ++i) {
    const float w = 1.0f / (float)(i + 2);
    hipMemsetAsync(acc, 0, (size_t)FEAT * sizeof(float), stream);

    auto sc = [&](const float* sF, int sc_, float* dF, int dc_,
                  const int* si, const int* di, int E) {
      scatter_kernel<<<((long long)E * 16 + 255) / 256, 256, 0, stream>>>(sF, sc_, dF, dc_, si, di, E);
    };
    // user dst: left=vu (v[:, :H]), right=pu (p[:, :H])
    sc(cur + vOff * EMB, 0,    acc + uOff * EMB, 0,    vu_s, vu_d, E_vu);
    sc(cur + pOff * EMB, 0,    acc + uOff * EMB, HALF, pu_s, pu_d, E_pu);
    // video dst: left=uv (u[:, :H]), right=tv (t[:, :H])
    sc(cur + uOff * EMB, 0,    acc + vOff * EMB, 0,    uv_s, uv_d, E_uv);
    sc(cur + tOff * EMB, 0,    acc + vOff * EMB, HALF, tv_s, tv_d, E_tv);
    // publisher dst: left=up (u[:, H:]), right=tp (t[:, H:])
    sc(cur + uOff * EMB, HALF, acc + pOff * EMB, 0,    up_s, up_d, E_up);
    sc(cur + tOff * EMB, HALF, acc + pOff * EMB, HALF, tp_s, tp_d, E_tp);
    // tag dst: left=vt (v[:, H:]), right=pt (p[:, H:])
    sc(cur + vOff * EMB, HALF, acc + tOff * EMB, 0,    vt_s, vt_d, E_vt);
    sc(cur + pOff * EMB, HALF, acc + tOff * EMB, HALF, pt_s, pt_d, E_pt);

    auto fin = [&](long long off, long long oL, long long oR, int n) {
      int stripes = n / 16;                        // full 16-row stripes (unguarded)
      if (stripes > 0) {
        int blocks = (stripes + 7) / 8;            // 256 threads = 8 wave32
        finalize_full_kernel<<<blocks, 256, 0, stream>>>(acc + off * EMB, cur + off * EMB,
                                                         out + off * EMB, inv + oL, inv + oR,
                                                         stripes, w);
      }
      int rowStart = stripes * 16;
      if (rowStart < n) {
        int nTail = (n - rowStart) * EMB;
        finalize_tail_kernel<<<(nTail + 255) / 256, 256, 0, stream>>>(
            acc + off * EMB, cur + off * EMB, out + off * EMB,
            inv + oL, inv + oR, rowStart, n, w);
      }
    };
    fin(uOff, o_vu, o_pu, NU);
    fin(vOff, o_uv, o_tv, NV);
    fin(pOff, o_up, o_tp, NP);
    fin(tOff, o_vt, o_pt, NT);
  }
}